// testModel_31104153158311
// MI455X (gfx1250) — compile-verified
//
#include <hip/hip_runtime.h>
#include <hip/hip_bf16.h>

typedef __attribute__((ext_vector_type(2))) float v2f;
typedef __attribute__((ext_vector_type(8))) float v8f;

#define D 20
#define NT 4
#define ET 6
#define PW 32   // padded column stride for LDS weight tiles (zeros in cols 20..31)

__device__ __forceinline__ v8f wmma4(v2f a, v2f b, v8f c) {
  // V_WMMA_F32_16X16X4_F32 : D(16x16 f32) = A(16x4 f32) * B(4x16 f32) + C
  return __builtin_amdgcn_wmma_f32_16x16x4_f32(
      /*neg_a=*/false, a, /*neg_b=*/false, b,
      /*c_mod=*/(short)0, c, /*reuse_a=*/false, /*reuse_b=*/false);
}

// ---------- ordered-int mapping for float atomicMax (exact, monotone) -------
__device__ __forceinline__ int f2key(float f) {
  int b = __float_as_int(f);
  return b >= 0 ? b : (b ^ 0x7FFFFFFF);
}
__device__ __forceinline__ float key2f(int k) {
  return __int_as_float(k >= 0 ? k : (k ^ 0x7FFFFFFF));
}

// ---------------------------------------------------------------------------
// k0: init z, denom, segmax
__global__ void init_kernel(float* __restrict__ z, float* __restrict__ denom,
                            int* __restrict__ segmax, int N) {
  int i = blockIdx.x * blockDim.x + threadIdx.x;
  if (i < N * D) z[i] = 0.0f;
  if (i < N) { denom[i] = 0.0f; segmax[i] = (int)0x80000000; }
}

// ---------------------------------------------------------------------------
// k1: fold edge hetero-linear into attention:  u[t] = W_t @ w3,  c[t] = b_t . w3
__global__ void prep_kernel(const float* __restrict__ eW, const float* __restrict__ eb,
                            const float* __restrict__ attW,
                            float* __restrict__ u, float* __restrict__ cvec) {
  int i = threadIdx.x;
  if (i < ET * D) {
    int t = i / D, k = i % D;
    float s = 0.0f;
    for (int d = 0; d < D; ++d) s += eW[t * D * D + k * D + d] * attW[2 * D + d];
    u[i] = s;
  }
  if (i < ET) {
    float s = 0.0f;
    for (int d = 0; d < D; ++d) s += eb[i * D + d] * attW[2 * D + d];
    cvec[i] = s;
  }
}

// ---------------------------------------------------------------------------
// k2: node hetero-linear via WMMA fp32. 16 nodes per wave; per-type row masking
// keeps mixed-type tiles exact. B tiles live in LDS padded to 32 cols (zeros),
// so every B fetch is an unconditional ds_load_2addr (no EXEC juggling).
__global__ __launch_bounds__(256) void node_linear_kernel(
    const float* __restrict__ x, const int* __restrict__ ntype,
    const float* __restrict__ nW, const float* __restrict__ nb,
    float* __restrict__ xn, int N) {
  __shared__ float sW[NT * D * PW];   // 2560 floats, cols 20..31 = 0
  __shared__ float snb[NT * PW];      // padded bias
  for (int i = threadIdx.x; i < NT * D * PW; i += blockDim.x) {
    int col = i & (PW - 1);
    int rk  = i >> 5;                 // (type*D + k)
    sW[i] = (col < D) ? nW[rk * D + col] : 0.0f;
  }
  for (int i = threadIdx.x; i < NT * PW; i += blockDim.x) {
    int col = i & (PW - 1);
    int t   = i >> 5;
    snb[i] = (col < D) ? nb[t * D + col] : 0.0f;
  }
  __syncthreads();

  int wave = threadIdx.x >> 5;
  int lane = threadIdx.x & 31;
  int base = (blockIdx.x * (blockDim.x >> 5) + wave) * 16;
  if (base >= N) return;

  int M = lane & 15;
  int hi = lane >> 4;
  int node = base + M;
  bool rv = node < N;
  int mytype = rv ? ntype[node] : -1;

  // A 16x4 layout: lanes 0-15 -> K={4t, 4t+1}; lanes 16-31 -> K={4t+2, 4t+3}
  v2f a[5];
  const float* xr = x + (size_t)node * D;
#pragma unroll
  for (int t = 0; t < 5; ++t) {
    int k0 = 4 * t + 2 * hi;
    a[t].x = rv ? xr[k0] : 0.0f;
    a[t].y = rv ? xr[k0 + 1] : 0.0f;
  }

  v8f c0 = {0.f, 0.f, 0.f, 0.f, 0.f, 0.f, 0.f, 0.f};
  v8f c1 = {0.f, 0.f, 0.f, 0.f, 0.f, 0.f, 0.f, 0.f};
  int n0 = M;        // cols 0..15
  int n1 = M + 16;   // cols 16..31 (20..31 are zero-padded)

#pragma unroll
  for (int kt = 0; kt < NT; ++kt) {
    const float* wb = sW + kt * D * PW;
#pragma unroll
    for (int t = 0; t < 5; ++t) {
      int krow = 4 * t + 2 * hi;     // krow, krow+1 <= 19
      v2f am;
      am.x = (mytype == kt) ? a[t].x : 0.0f;
      am.y = (mytype == kt) ? a[t].y : 0.0f;
      v2f b0; b0.x = wb[krow * PW + n0]; b0.y = wb[(krow + 1) * PW + n0];
      c0 = wmma4(am, b0, c0);
      v2f b1; b1.x = wb[krow * PW + n1]; b1.y = wb[(krow + 1) * PW + n1];
      c1 = wmma4(am, b1, c1);
    }
  }

  // C/D layout: lanes 0-15 vgpr v -> row M=v ; lanes 16-31 -> row M=8+v
#pragma unroll
  for (int v = 0; v < 8; ++v) {
    int rr = hi * 8 + v;
    int row = base + rr;
    int rt = __shfl(mytype, rr, 32);          // type of row rr (from lane rr)
    if (row < N) {
      xn[(size_t)row * D + n0] = c0[v] + snb[rt * PW + n0];
      if (n1 < D) xn[(size_t)row * D + n1] = c1[v] + snb[rt * PW + n1];
    }
  }
}

// ---------------------------------------------------------------------------
// k3: per-node attention dots  s1[n] = xn[n].w1 (dst term), s2[n] = xn[n].w2 (src term)
__global__ void sdots_kernel(const float* __restrict__ xn, const float* __restrict__ attW,
                             float* __restrict__ s1, float* __restrict__ s2, int N) {
  __shared__ float w[2 * D];
  if (threadIdx.x < 2 * D) w[threadIdx.x] = attW[threadIdx.x];
  __syncthreads();
  int n = blockIdx.x * blockDim.x + threadIdx.x;
  if (n >= N) return;
  const float4* xr = (const float4*)(xn + (size_t)n * D);
  float a = 0.0f, b = 0.0f;
#pragma unroll
  for (int q = 0; q < 5; ++q) {
    float4 v = xr[q];
    a += v.x * w[q * 4 + 0] + v.y * w[q * 4 + 1] + v.z * w[q * 4 + 2] + v.w * w[q * 4 + 3];
    b += v.x * w[D + q * 4 + 0] + v.y * w[D + q * 4 + 1] + v.z * w[D + q * 4 + 2] + v.w * w[D + q * 4 + 3];
  }
  s1[n] = a;
  s2[n] = b;
}

// ---------------------------------------------------------------------------
// k4: edge pass 1 — raw attention logit + segment max (ordered-int atomicMax)
__global__ void edge_alpha_kernel(const int* __restrict__ ei, const int* __restrict__ etype,
                                  const float* __restrict__ attr,
                                  const float* __restrict__ s1, const float* __restrict__ s2,
                                  const float* __restrict__ u, const float* __restrict__ cvec,
                                  const float* __restrict__ attB,
                                  float* __restrict__ alpha, int* __restrict__ segmax, int E) {
  __shared__ float su[ET * D];
  __shared__ float sc[ET];
  if (threadIdx.x < ET * D) su[threadIdx.x] = u[threadIdx.x];
  if (threadIdx.x < ET) sc[threadIdx.x] = cvec[threadIdx.x];
  __syncthreads();
  int e = blockIdx.x * blockDim.x + threadIdx.x;
  if (e >= E) return;
  int src = ei[e];
  int dst = ei[E + e];
  int t = etype[e];
  const float4* ar = (const float4*)(attr + (size_t)e * D);
  const float* ut = su + t * D;
  float acc = 0.0f;
#pragma unroll
  for (int q = 0; q < 5; ++q) {
    float4 v = ar[q];
    acc += v.x * ut[q * 4 + 0] + v.y * ut[q * 4 + 1] + v.z * ut[q * 4 + 2] + v.w * ut[q * 4 + 3];
  }
  float a = attB[0] + s1[dst] + s2[src] + sc[t] + acc;
  a = a >= 0.0f ? a : 0.2f * a;   // leaky_relu(0.2)
  alpha[e] = a;
  atomicMax(&segmax[dst], f2key(a));
}

// ---------------------------------------------------------------------------
// k5: edge pass 2 — exp(a - max) and denom accumulation
__global__ void edge_exp_kernel(const int* __restrict__ ei, float* __restrict__ alpha,
                                const int* __restrict__ segmax, float* __restrict__ denom, int E) {
  int e = blockIdx.x * blockDim.x + threadIdx.x;
  if (e >= E) return;
  int dst = ei[E + e];
  float m = key2f(segmax[dst]);
  float ex = expf(alpha[e] - m);
  alpha[e] = ex;
  atomicAdd(&denom[dst], ex);
}

// ---------------------------------------------------------------------------
// k6: edge pass 3 — z[dst] += (alpha/denom) * xn[src]
__global__ void edge_scatter_kernel(const int* __restrict__ ei, const float* __restrict__ alpha,
                                    const float* __restrict__ denom, const float* __restrict__ xn,
                                    float* __restrict__ z, int E) {
  int e = blockIdx.x * blockDim.x + threadIdx.x;
  if (e >= E) return;
  int src = ei[e];
  int dst = ei[E + e];
  float coef = alpha[e] / (denom[dst] + 1e-16f);
  const float4* xr = (const float4*)(xn + (size_t)src * D);
  float* zr = z + (size_t)dst * D;
#pragma unroll
  for (int q = 0; q < 5; ++q) {
    float4 v = xr[q];
    atomicAdd(zr + q * 4 + 0, v.x * coef);
    atomicAdd(zr + q * 4 + 1, v.y * coef);
    atomicAdd(zr + q * 4 + 2, v.z * coef);
    atomicAdd(zr + q * 4 + 3, v.w * coef);
  }
}

// ---------------------------------------------------------------------------
// k7: decoder via WMMA fp32. 16 labels per wave:
//   h = relu([z[row], z[col]] (16x40) @ W1 (40x20) + b1); pred = h @ W2 + b2
// W1/W2/b1 padded to 32 cols with zeros -> branch-free WMMA stream.
__global__ __launch_bounds__(256) void decoder_kernel(
    const float* __restrict__ z, const int* __restrict__ eli,
    const float* __restrict__ W1, const float* __restrict__ b1,
    const float* __restrict__ W2, const float* __restrict__ b2,
    float* __restrict__ pred, int L) {
  __shared__ float sW1[2 * D * PW];  // 40 x 32, cols 20..31 = 0
  __shared__ float sW2[PW];
  __shared__ float sb1[PW];
  __shared__ float red[8][16];
  for (int i = threadIdx.x; i < 2 * D * PW; i += blockDim.x) {
    int col = i & (PW - 1);
    int k   = i >> 5;
    sW1[i] = (col < D) ? W1[k * D + col] : 0.0f;
  }
  if (threadIdx.x < PW) {
    int col = threadIdx.x;
    sW2[col] = (col < D) ? W2[col] : 0.0f;
    sb1[col] = (col < D) ? b1[col] : 0.0f;
  }
  __syncthreads();

  int wave = threadIdx.x >> 5;
  int lane = threadIdx.x & 31;
  int base = (blockIdx.x * (blockDim.x >> 5) + wave) * 16;
  int M = lane & 15;
  int hi = lane >> 4;
  int lab = base + M;
  bool lv = lab < L;
  int r = lv ? eli[lab] : 0;
  int c = lv ? eli[L + lab] : 0;
  const float* zr = z + (size_t)r * D;
  const float* zc = z + (size_t)c * D;

  v2f a[10];
#pragma unroll
  for (int t = 0; t < 10; ++t) {
    int k0 = 4 * t + 2 * hi;                 // k0 even; {k0,k0+1} on same side of 20
    a[t].x = (k0 < D) ? zr[k0] : zc[k0 - D];
    a[t].y = (k0 + 1 < D) ? zr[k0 + 1] : zc[k0 + 1 - D];
  }

  v8f c0 = {0.f, 0.f, 0.f, 0.f, 0.f, 0.f, 0.f, 0.f};
  v8f c1 = {0.f, 0.f, 0.f, 0.f, 0.f, 0.f, 0.f, 0.f};
  int n0 = M, n1 = M + 16;
#pragma unroll
  for (int t = 0; t < 10; ++t) {
    int krow = 4 * t + 2 * hi;
    v2f b0; b0.x = sW1[krow * PW + n0]; b0.y = sW1[(krow + 1) * PW + n0];
    c0 = wmma4(a[t], b0, c0);
    v2f b1v; b1v.x = sW1[krow * PW + n1]; b1v.y = sW1[(krow + 1) * PW + n1];
    c1 = wmma4(a[t], b1v, c1);
  }

  float w2a = sW2[n0], b1a = sb1[n0];
  float w2b = sW2[n1], b1b = sb1[n1];     // zeros for padded cols
  float pv[8];
#pragma unroll
  for (int v = 0; v < 8; ++v) {
    float h0 = c0[v] + b1a; h0 = h0 > 0.0f ? h0 : 0.0f;
    float h1 = c1[v] + b1b; h1 = h1 > 0.0f ? h1 : 0.0f;   // padded: relu(0)=0
    pv[v] = h0 * w2a + h1 * w2b;                          // padded: *0
  }
  // reduce each row across the 16 lanes of each half (xor masks stay in-half)
#pragma unroll
  for (int v = 0; v < 8; ++v) {
    float s = pv[v];
    s += __shfl_xor(s, 1, 32);
    s += __shfl_xor(s, 2, 32);
    s += __shfl_xor(s, 4, 32);
    s += __shfl_xor(s, 8, 32);
    pv[v] = s;
  }
  if ((lane & 15) == 0) {
#pragma unroll
    for (int v = 0; v < 8; ++v) red[wave][hi * 8 + v] = pv[v];
  }
  __syncthreads();
  if (lane < 16 && lab < L) pred[lab] = red[wave][lane] + b2[0];
}

// ---------------------------------------------------------------------------
extern "C" void kernel_launch(void* const* d_in, const int* in_sizes, int n_in,
                              void* d_out, int out_size, void* d_ws, size_t ws_size,
                              hipStream_t stream) {
  const float* x    = (const float*)d_in[0];
  const int*   ei   = (const int*)d_in[1];
  const int*   ntyp = (const int*)d_in[2];
  const float* attr = (const float*)d_in[3];
  const int*   etyp = (const int*)d_in[4];
  const int*   eli  = (const int*)d_in[5];
  const float* nW   = (const float*)d_in[6];
  const float* nb   = (const float*)d_in[7];
  const float* eW   = (const float*)d_in[8];
  const float* eb   = (const float*)d_in[9];
  const float* attW = (const float*)d_in[10];
  const float* attB = (const float*)d_in[11];
  const float* dW1  = (const float*)d_in[12];
  const float* db1  = (const float*)d_in[13];
  const float* dW2  = (const float*)d_in[14];
  const float* db2  = (const float*)d_in[15];

  const int N = in_sizes[2];
  const int E = in_sizes[4];
  const int L = in_sizes[5] / 2;

  float* pred = (float*)d_out;
  float* z    = (float*)d_out + L;          // [N, D] second output, written in place

  float* ws     = (float*)d_ws;
  float* xn     = ws;                        // N*D
  float* alpha  = xn + (size_t)N * D;        // E
  float* s1     = alpha + E;                 // N
  float* s2     = s1 + N;                    // N
  float* denom  = s2 + N;                    // N
  int*   segmax = (int*)(denom + N);         // N
  float* u      = (float*)(segmax + N);      // ET*D (padded to 128)
  float* cvec   = u + 128;                   // ET

  const int B = 256;
  init_kernel<<<(N * D + B - 1) / B, B, 0, stream>>>(z, denom, segmax, N);
  prep_kernel<<<1, 128, 0, stream>>>(eW, eb, attW, u, cvec);
  node_linear_kernel<<<(N + 127) / 128, B, 0, stream>>>(x, ntyp, nW, nb, xn, N);
  sdots_kernel<<<(N + B - 1) / B, B, 0, stream>>>(xn, attW, s1, s2, N);
  edge_alpha_kernel<<<(E + B - 1) / B, B, 0, stream>>>(ei, etyp, attr, s1, s2, u, cvec,
                                                       attB, alpha, segmax, E);
  edge_exp_kernel<<<(E + B - 1) / B, B, 0, stream>>>(ei, alpha, segmax, denom, E);
  edge_scatter_kernel<<<(E + B - 1) / B, B, 0, stream>>>(ei, alpha, denom, xn, z, E);
  decoder_kernel<<<(L + 127) / 128, B, 0, stream>>>(z, eli, dW1, db1, dW2, db2, pred, L);
}